// YOLOV3HeadWraper_1202590843779
// MI455X (gfx1250) — compile-verified
//
#include <hip/hip_runtime.h>
#include <stdint.h>

#define NUM_CLASSES   80
#define BATCH         32
#define AT_TOTAL      22743       // 1083 + 4332 + 17328
#define NPRE          1000
#define NTOPK         1000
#define MAXDET        100
#define CONF_THR      0.005f
#define SCORE_THR     0.05f
#define IOU_THR       0.45f
#define CLS_OFF       4096.0f
#define NSCORES       240000      // 3000 * 80 per image

typedef __attribute__((ext_vector_type(2))) float v2f;
typedef __attribute__((ext_vector_type(8))) float v8f;
typedef unsigned int u32x4 __attribute__((ext_vector_type(4)));
typedef int i32x8 __attribute__((ext_vector_type(8)));
typedef int i32x4 __attribute__((ext_vector_type(4)));

__constant__ int   c_H[3]      = {19, 38, 76};
__constant__ int   c_HW[3]     = {361, 1444, 5776};
__constant__ float c_stride[3] = {32.f, 16.f, 8.f};
__constant__ int   c_loff[3]   = {0, 1083, 5415};
__constant__ int   c_lcnt[3]   = {1083, 4332, 17328};
__constant__ float c_bw[3][3]  = {{116.f,156.f,373.f},{30.f,62.f,59.f},{10.f,16.f,33.f}};
__constant__ float c_bh[3][3]  = {{90.f,198.f,326.f},{61.f,45.f,119.f},{13.f,30.f,23.f}};

__device__ __forceinline__ float sigmoidf(float x) { return 1.0f / (1.0f + expf(-x)); }
__device__ __forceinline__ unsigned orderKey(float f) {
    unsigned u = __float_as_uint(f);
    return (u & 0x80000000u) ? ~u : (u | 0x80000000u);
}
__device__ __forceinline__ float unorderKey(unsigned k) {
    unsigned u = (k & 0x80000000u) ? (k ^ 0x80000000u) : ~k;
    return __uint_as_float(u);
}

// ---------------------------------------------------------------------------
// Kernel 1: objectness sigmoid for every anchor of every level (streaming).
// ---------------------------------------------------------------------------
__global__ void decode_conf_kernel(const float* __restrict__ p0,
                                   const float* __restrict__ p1,
                                   const float* __restrict__ p2,
                                   float* __restrict__ conf_all, int total) {
    int gid = blockIdx.x * blockDim.x + threadIdx.x;
    if (gid >= total) return;
    int b = gid / AT_TOTAL;
    int a = gid % AT_TOTAL;
    int lvl = (a < 1083) ? 0 : (a < 5415 ? 1 : 2);
    int ai  = a - c_loff[lvl];
    const float* pm = (lvl == 0) ? p0 : (lvl == 1 ? p1 : p2);
    int H = c_H[lvl], HW = c_HW[lvl];
    int k = ai % 3, cell = ai / 3;
    int x = cell % H, y = cell / H;
    const float* src = pm + (size_t)(b * 255 + k * 85 + 4) * HW + (size_t)y * H + x;
    __builtin_prefetch(src, 0, 0);   // gfx1250: global_prefetch_b8
    conf_all[gid] = sigmoidf(*src);
}

// ---------------------------------------------------------------------------
// Bitonic sort (descending) of packed (key<<32 | idx) values in LDS.
// ---------------------------------------------------------------------------
__device__ void bitonic_sort_desc(unsigned long long* sm, int P, int tid, int nthr) {
    for (int k = 2; k <= P; k <<= 1) {
        for (int j = k >> 1; j > 0; j >>= 1) {
            for (int i = tid; i < P; i += nthr) {
                int ij = i ^ j;
                if (ij > i) {
                    bool up = ((i & k) == 0);
                    unsigned long long a = sm[i], b = sm[ij];
                    bool sw = up ? (a < b) : (a > b);   // descending overall
                    if (sw) { sm[i] = b; sm[ij] = a; }
                }
            }
            __syncthreads();
        }
    }
}

// ---------------------------------------------------------------------------
// Kernel 2: exact per-level top-1000 via full bitonic sort in 320KB LDS.
// One block per image; dynamic LDS = P*8 bytes (up to 256 KB for level 2).
// ---------------------------------------------------------------------------
__global__ void topk_level_kernel(const float* __restrict__ conf_all,
                                  int level_off, int A, int P, int lvl,
                                  int* __restrict__ sel_idx,
                                  float* __restrict__ sel_conf) {
    extern __shared__ char dynsm[];
    unsigned long long* sm = (unsigned long long*)dynsm;
    int b = blockIdx.x, tid = threadIdx.x, nthr = blockDim.x;
    const float* conf = conf_all + (size_t)b * AT_TOTAL + level_off;

    for (int i = tid; i < P; i += nthr) {
        unsigned long long v = 0ull;
        if (i < A) v = ((unsigned long long)orderKey(conf[i]) << 32) | (unsigned)i;
        sm[i] = v;
    }
    __syncthreads();
    bitonic_sort_desc(sm, P, tid, nthr);

    if (tid < NPRE) {
        unsigned long long v = sm[tid];
        int idx = (int)(v & 0xFFFFFFFFull);
        float cf = conf[idx];
        cf = cf * ((cf >= CONF_THR) ? 1.0f : 0.0f);
        int o = (b * 3 + lvl) * NPRE + tid;
        sel_idx[o]  = idx;
        sel_conf[o] = cf;
    }
}

// ---------------------------------------------------------------------------
// Kernel 3: decode bbox + 80 class scores for the 3000 selected anchors/image.
// ---------------------------------------------------------------------------
__global__ void gather_kernel(const float* __restrict__ p0,
                              const float* __restrict__ p1,
                              const float* __restrict__ p2,
                              const int* __restrict__ sel_idx,
                              const float* __restrict__ sel_conf,
                              float* __restrict__ cand_bbox,
                              float* __restrict__ cand_score) {
    int gid = blockIdx.x * blockDim.x + threadIdx.x;
    if (gid >= BATCH * 3000) return;
    int b   = gid / 3000;
    int r3  = gid % 3000;
    int lvl = r3 / NPRE;
    int r   = r3 % NPRE;
    int o   = (b * 3 + lvl) * NPRE + r;
    int ai  = sel_idx[o];
    float conf = sel_conf[o];

    const float* pm = (lvl == 0) ? p0 : (lvl == 1 ? p1 : p2);
    int H = c_H[lvl], HW = c_HW[lvl];
    float stride = c_stride[lvl];
    int k = ai % 3, cell = ai / 3;
    int x = cell % H, y = cell / H;
    size_t base = (size_t)(b * 255 + k * 85) * HW + (size_t)y * H + x;

    float t0 = pm[base + 0 * (size_t)HW];
    float t1 = pm[base + 1 * (size_t)HW];
    float t2 = pm[base + 2 * (size_t)HW];
    float t3 = pm[base + 3 * (size_t)HW];
    float sx = sigmoidf(sigmoidf(t0));
    float sy = sigmoidf(sigmoidf(t1));
    float acx = (float)x * stride + 0.5f * stride;
    float acy = (float)y * stride + 0.5f * stride;
    float cx = (sx - 0.5f) * stride + acx;
    float cy = (sy - 0.5f) * stride + acy;
    float w = expf(t2) * c_bw[lvl][k];
    float h = expf(t3) * c_bh[lvl][k];

    float* bb = cand_bbox + (size_t)gid * 4;
    bb[0] = cx - 0.5f * w; bb[1] = cy - 0.5f * h;
    bb[2] = cx + 0.5f * w; bb[3] = cy + 0.5f * h;

    float* cs = cand_score + (size_t)gid * NUM_CLASSES;
    for (int c = 0; c < NUM_CLASSES; ++c) {
        float t = pm[base + (size_t)(5 + c) * HW];
        cs[c] = conf * sigmoidf(t);
    }
}

// ---------------------------------------------------------------------------
// Kernel 4: per-image top-1000 of 240000 scores: 3-pass radix select on
// order-preserving float keys, compaction, then LDS bitonic sort of 1024.
// ---------------------------------------------------------------------------
__global__ void select_scores_kernel(const float* __restrict__ cand_score,
                                     float* __restrict__ sel2_score,
                                     int*   __restrict__ sel2_idx) {
    __shared__ int hist[2048];
    __shared__ unsigned long long pairs[1024];
    __shared__ int s_bin, s_need, s_cntGt, s_cntEq;

    int b = blockIdx.x, tid = threadIdx.x, nthr = blockDim.x;
    const float* sc = cand_score + (size_t)b * NSCORES;

    unsigned prefix = 0, knownMask = 0;
    int need = NTOPK;
    const int shifts[3] = {21, 10, 0};
    const int nbits[3]  = {11, 11, 10};

    for (int p = 0; p < 3; ++p) {
        int nb = 1 << nbits[p], sh = shifts[p];
        for (int t = tid; t < 2048; t += nthr) hist[t] = 0;
        __syncthreads();
        for (int i = tid; i < NSCORES; i += nthr) {
            float s = sc[i]; s = (s > SCORE_THR) ? s : 0.0f;
            unsigned key = orderKey(s);
            if ((key & knownMask) == prefix)
                atomicAdd(&hist[(key >> sh) & (nb - 1)], 1);
        }
        __syncthreads();
        if (tid == 0) {
            int cum = 0, t;
            for (t = nb - 1; t > 0; --t) {
                if (cum + hist[t] >= need) break;
                cum += hist[t];
            }
            s_bin = t; s_need = need - cum;
        }
        __syncthreads();
        prefix |= ((unsigned)s_bin) << sh;
        need = s_need;
        knownMask |= ((unsigned)(nb - 1)) << sh;
        __syncthreads();
    }

    // prefix == exact 32-bit threshold key; need == # equal elements to take.
    int totalGt = NTOPK - need;
    if (tid == 0) { s_cntGt = 0; s_cntEq = 0; }
    for (int t = tid; t < 1024; t += nthr) pairs[t] = 0ull;
    __syncthreads();

    for (int i = tid; i < NSCORES; i += nthr) {
        float s = sc[i]; s = (s > SCORE_THR) ? s : 0.0f;
        unsigned key = orderKey(s);
        if (key > prefix) {
            int p = atomicAdd(&s_cntGt, 1);
            if (p < 1024) pairs[p] = ((unsigned long long)key << 32) | (unsigned)i;
        } else if (key == prefix) {
            int p = atomicAdd(&s_cntEq, 1);
            if (p < need && (totalGt + p) < 1024)
                pairs[totalGt + p] = ((unsigned long long)key << 32) | (unsigned)i;
        }
    }
    __syncthreads();
    bitonic_sort_desc(pairs, 1024, tid, nthr);

    if (tid < NTOPK) {
        unsigned long long v = pairs[tid];
        sel2_score[(size_t)b * NTOPK + tid] = unorderKey((unsigned)(v >> 32));
        sel2_idx  [(size_t)b * NTOPK + tid] = (int)(v & 0xFFFFFFFFull);
    }
}

// ---------------------------------------------------------------------------
// Kernel 5: NMS. Boxes + full 1000x1000 suppression bitmask live in LDS
// (~153 KB, only possible with CDNA5's 320 KB/WGP). The contiguous score
// vector is DMA'd into LDS by the Tensor Data Mover (TENSOR_LOAD_TO_LDS,
// tracked by TENSORcnt). Union areas per 16x16 tile come from
// V_WMMA_F32_16X16X4_F32 (outer sum area_i + area_j), intersections from
// VALU min/max; then a serial bitmask suppression sweep.
// ---------------------------------------------------------------------------
#define LDS_ADJ    0
#define LDS_X1     128000
#define LDS_Y1     132000
#define LDS_X2     136000
#define LDS_Y2     140000
#define LDS_AREA   144000
#define LDS_SCR    148000
#define LDS_CLS    152000
#define LDS_KEEP   156000
#define LDS_TOTAL  156160

__global__ void nms_kernel(const float* __restrict__ cand_bbox,
                           const float* __restrict__ sel2_score,
                           const int*   __restrict__ sel2_idx,
                           float* __restrict__ out) {
    extern __shared__ char dynsm[];
    unsigned* s_adj  = (unsigned*)(dynsm + LDS_ADJ);   // 1000*32 u32
    float* s_x1   = (float*)(dynsm + LDS_X1);
    float* s_y1   = (float*)(dynsm + LDS_Y1);
    float* s_x2   = (float*)(dynsm + LDS_X2);
    float* s_y2   = (float*)(dynsm + LDS_Y2);
    float* s_area = (float*)(dynsm + LDS_AREA);
    float* s_scr  = (float*)(dynsm + LDS_SCR);
    int*   s_cls  = (int*)  (dynsm + LDS_CLS);
    unsigned* s_keep = (unsigned*)(dynsm + LDS_KEEP);  // 32 u32

    int b = blockIdx.x, tid = threadIdx.x, nthr = blockDim.x;

    // --- TDM: DMA the 1000 contiguous fp32 scores into LDS (wave 0 only). ---
    if (tid < 32) {
        unsigned long long ga =
            (unsigned long long)(uintptr_t)(sel2_score + (size_t)b * NTOPK);
        u32x4 g0;
        g0[0] = 1u;                               // count=1, user mode, no gather
        g0[1] = (unsigned)LDS_SCR;                // lds_addr (dynamic LDS base = 0)
        g0[2] = (unsigned)ga;                     // global_addr[31:0]
        g0[3] = ((unsigned)(ga >> 32) & 0x01FFFFFFu) | 0x80000000u; // addr[56:32] | type=2
        i32x8 g1;
        g1[0] = (2 << 16);                        // workgroup_mask=0, data_size=4B
        g1[1] = (int)((NTOPK & 0xFFFF) << 16);    // tensor_dim0[15:0] in [31:16]
        g1[2] = (int)(1u << 16);                  // tensor_dim0 hi=0; tensor_dim1=1
        g1[3] = (int)(NTOPK << 16);               // tensor_dim1 hi=0; tile_dim0=1000
        g1[4] = 0;                                // tile_dim1=0, tile_dim2=0
        g1[5] = NTOPK;                            // tensor_dim0_stride lo
        g1[6] = 0;
        g1[7] = 0;
        i32x4 z4 = {0, 0, 0, 0};                  // groups 2/3 unused (<=2D tensor)
        i32x8 z8 = {0, 0, 0, 0, 0, 0, 0, 0};
        __builtin_amdgcn_tensor_load_to_lds(g0, g1, z4, z4, z8, 0);
        __builtin_amdgcn_s_wait_tensorcnt((short)0);
    }

    // Load candidates, apply class offset (matches reference semantics).
    for (int j = tid; j < NTOPK; j += nthr) {
        int fi = sel2_idx[(size_t)b * NTOPK + j];
        int q  = fi / NUM_CLASSES;
        int c  = fi % NUM_CLASSES;
        const float* bb = cand_bbox + ((size_t)b * 3000 + q) * 4;
        float off = (float)c * CLS_OFF;
        float x1 = bb[0] + off, y1 = bb[1] + off;
        float x2 = bb[2] + off, y2 = bb[3] + off;
        s_x1[j] = x1; s_y1[j] = y1; s_x2[j] = x2; s_y2[j] = y2;
        s_area[j] = fmaxf(x2 - x1, 0.0f) * fmaxf(y2 - y1, 0.0f);
        s_cls[j]  = c;
    }
    for (int t = tid; t < NTOPK * 32; t += nthr) s_adj[t] = 0u;
    __syncthreads();

    // Build suppression adjacency. 63x63 tiles of 16x16; one tile per wave.
    int lane  = tid & 31;
    int wave  = tid >> 5;
    int nwav  = nthr >> 5;
    v8f czero = {};
    for (int t = wave; t < 63 * 63; t += nwav) {
        int ti = t / 63, tj = t % 63;
        int i0 = ti * 16, j0 = tj * 16;

        float areaI = 0.f, oneI = 0.f, oneJ = 0.f, areaJ = 0.f;
        if (lane < 16) {
            int ii = i0 + lane; areaI = (ii < NTOPK) ? s_area[ii] : 0.f;
            int jj = j0 + lane; areaJ = (jj < NTOPK) ? s_area[jj] : 0.f;
            oneI = 1.0f; oneJ = 1.0f;
        }
        // A (16x4): col K0 = area_i, col K1 = 1      (lanes>=16 -> K2,K3 = 0)
        // B (4x16): row K0 = 1,      row K1 = area_j (lanes>=16 -> K2,K3 = 0)
        v2f A;  A[0]  = areaI; A[1]  = oneI;
        v2f Bv; Bv[0] = oneJ;  Bv[1] = areaJ;
        v8f d = __builtin_amdgcn_wmma_f32_16x16x4_f32(
            false, A, false, Bv, (short)0, czero, false, false);
        // d[m][n] = area_i(m) + area_j(n)

        int nj = j0 + (lane & 15);
        int mb = (lane >> 4) << 3;
        for (int r = 0; r < 8; ++r) {
            int mi = i0 + mb + r;
            if (mi < NTOPK && nj < NTOPK && mi != nj) {
                float ix1 = fmaxf(s_x1[mi], s_x1[nj]);
                float iy1 = fmaxf(s_y1[mi], s_y1[nj]);
                float ix2 = fminf(s_x2[mi], s_x2[nj]);
                float iy2 = fminf(s_y2[mi], s_y2[nj]);
                float inter = fmaxf(ix2 - ix1, 0.f) * fmaxf(iy2 - iy1, 0.f);
                float uni = d[r] - inter;              // WMMA area sum - inter
                if (inter > IOU_THR * fmaxf(uni, 1e-6f))
                    atomicOr(&s_adj[mi * 32 + (nj >> 5)], 1u << (nj & 31));
            }
        }
    }

    // keep init: score > 0
    __syncthreads();
    if (tid < 32) {
        unsigned k = 0;
        for (int bit = 0; bit < 32; ++bit) {
            int j = tid * 32 + bit;
            if (j < NTOPK && s_scr[j] > 0.0f) k |= (1u << bit);
        }
        s_keep[tid] = k;
    }
    __syncthreads();

    // Greedy sweep: if keep[i], clear keep[j] for j>i with adj bit set.
    for (int i = 0; i < NTOPK; ++i) {
        bool ki = (s_keep[i >> 5] >> (i & 31)) & 1u;
        __syncthreads();
        if (ki && tid < 32) {
            unsigned m = s_adj[i * 32 + tid];
            int iw = i >> 5;
            if (tid < iw) m = 0u;
            else if (tid == iw) {
                unsigned le = ((i & 31) == 31) ? 0xFFFFFFFFu
                                              : ((1u << ((i & 31) + 1)) - 1u);
                m &= ~le;
            }
            s_keep[tid] &= ~m;
        }
        __syncthreads();
    }

    // Emit top-100 (already score-sorted); outputs flattened in return order.
    if (tid == 0) {
        float* outNum = out + b;
        float* outBox = out + BATCH + (size_t)b * MAXDET * 4;
        float* outS   = out + BATCH + (size_t)BATCH * MAXDET * 4 + (size_t)b * MAXDET;
        float* outC   = outS + (size_t)BATCH * MAXDET;
        int cnt = 0;
        for (int j = 0; j < NTOPK && cnt < MAXDET; ++j) {
            if (((s_keep[j >> 5] >> (j & 31)) & 1u) && s_scr[j] > 0.0f) {
                float off = (float)s_cls[j] * CLS_OFF;
                outBox[cnt * 4 + 0] = s_x1[j] - off;
                outBox[cnt * 4 + 1] = s_y1[j] - off;
                outBox[cnt * 4 + 2] = s_x2[j] - off;
                outBox[cnt * 4 + 3] = s_y2[j] - off;
                outS[cnt] = s_scr[j];
                outC[cnt] = (float)s_cls[j];
                ++cnt;
            }
        }
        outNum[0] = (float)cnt;
        for (int m = cnt; m < MAXDET; ++m) {
            outBox[m * 4 + 0] = 0.f; outBox[m * 4 + 1] = 0.f;
            outBox[m * 4 + 2] = 0.f; outBox[m * 4 + 3] = 0.f;
            outS[m] = 0.f; outC[m] = -1.f;
        }
    }
}

// ---------------------------------------------------------------------------
extern "C" void kernel_launch(void* const* d_in, const int* in_sizes, int n_in,
                              void* d_out, int out_size, void* d_ws, size_t ws_size,
                              hipStream_t stream) {
    const float* p0 = (const float*)d_in[0];
    const float* p1 = (const float*)d_in[1];
    const float* p2 = (const float*)d_in[2];
    float* out = (float*)d_out;

    char* ws = (char*)d_ws;
    size_t off = 0;
    auto alloc = [&](size_t bytes) {
        void* p = ws + off;
        off += (bytes + 255) & ~(size_t)255;
        return p;
    };
    float* conf_all   = (float*)alloc((size_t)BATCH * AT_TOTAL * 4);
    int*   sel_idx    = (int*)  alloc((size_t)BATCH * 3 * NPRE * 4);
    float* sel_conf   = (float*)alloc((size_t)BATCH * 3 * NPRE * 4);
    float* cand_bbox  = (float*)alloc((size_t)BATCH * 3000 * 4 * 4);
    float* cand_score = (float*)alloc((size_t)BATCH * NSCORES * 4);
    float* sel2_score = (float*)alloc((size_t)BATCH * NTOPK * 4);
    int*   sel2_idx   = (int*)  alloc((size_t)BATCH * NTOPK * 4);

    int total = BATCH * AT_TOTAL;
    decode_conf_kernel<<<(total + 255) / 256, 256, 0, stream>>>(p0, p1, p2, conf_all, total);

    const int loff[3] = {0, 1083, 5415};
    const int lcnt[3] = {1083, 4332, 17328};
    const int lpad[3] = {2048, 8192, 32768};
    for (int lvl = 0; lvl < 3; ++lvl) {
        size_t lds = (size_t)lpad[lvl] * 8;   // up to 256 KB of CDNA5 LDS
        topk_level_kernel<<<BATCH, 1024, lds, stream>>>(
            conf_all, loff[lvl], lcnt[lvl], lpad[lvl], lvl, sel_idx, sel_conf);
    }

    gather_kernel<<<(BATCH * 3000 + 255) / 256, 256, 0, stream>>>(
        p0, p1, p2, sel_idx, sel_conf, cand_bbox, cand_score);

    select_scores_kernel<<<BATCH, 1024, 0, stream>>>(cand_score, sel2_score, sel2_idx);

    nms_kernel<<<BATCH, 1024, LDS_TOTAL, stream>>>(cand_bbox, sel2_score, sel2_idx, out);
}